// DiFGridEncoder_89215060672779
// MI455X (gfx1250) — compile-verified
//
#include <hip/hip_runtime.h>
#include <cstdint>
#include <cstddef>

typedef float v4f __attribute__((ext_vector_type(4)));

// ---------------------------------------------------------------------------
// Transpose one grid from (C, N) channel-major to (N, C) channel-last.
// N = reso^3. Fully coalesced on both sides via an LDS tile.
// ---------------------------------------------------------------------------
template<int C>
__global__ __launch_bounds__(256) void transpose_kernel(
    const float* __restrict__ src, float* __restrict__ dst, int N) {
  constexpr int S = 256 / C;            // spatial elements per block
  __shared__ float tile[C][S + 1];      // +1 pad to spread LDS banks
  const int tid = threadIdx.x;
  const int sBase = blockIdx.x * S;
  {
    const int c = tid / S;
    const int s = tid - c * S;          // consecutive lanes -> consecutive s
    if (sBase + s < N)
      tile[c][s] = __builtin_nontemporal_load(&src[(size_t)c * N + (sBase + s)]);
  }
  __syncthreads();
  {
    const int s = tid / C;
    const int c = tid - s * C;          // write index = sBase*C + tid (contiguous)
    if (sBase + s < N)
      dst[(size_t)(sBase + s) * C + c] = tile[c][s];
  }
}

// ---------------------------------------------------------------------------
// Exact sawtooth remainders for all six periods with only ONE real fmodf.
//   scale_R = 2/R;  p_R = fmod(x+1, scale_R)
// R=32: scale is 2^-4, remainder computed exactly with floor+mul+sub.
// R=64,128 derive from p32; R=96,192 derive from p48 (fmodf) — each via the
// exact identity fmod(v,h) = fmod(fmod(v,2h),h) with a Sterbenz-exact p-h.
// ---------------------------------------------------------------------------
__device__ __forceinline__ void sawtooth_all(float x,
    float& p32, float& p48, float& p64, float& p96, float& p128, float& p192) {
#pragma clang fp contract(off)
  const float v = x + 1.0f;                       // in [0, 2]
  {
    const float q = floorf(v * 16.0f);            // exact (v*16 <= 32)
    p32 = v - q * 0.0625f;                        // exact fmod(v, 2^-4)
  }
  const float s64 = 2.0f / 64.0f, s128 = 2.0f / 128.0f;
  p64  = (p32 < s64)  ? p32 : (p32 - s64);        // exact
  p128 = (p64 < s128) ? p64 : (p64 - s128);       // exact
  const float s48 = 2.0f / 48.0f, s96 = 2.0f / 96.0f, s192 = 2.0f / 192.0f;
  p48  = fmodf(v, s48);                           // exact by definition
  p96  = (p48 < s96)  ? p48 : (p48 - s96);        // exact
  p192 = (p96 < s192) ? p96 : (p96 - s192);       // exact
}

// ---------------------------------------------------------------------------
// Per-grid trilinear gather. Takes the precomputed sawtooth remainders.
// For power-of-two R the division by (scale*0.5) is an exact multiply by R;
// for R in {48,96,192} the true IEEE division is kept to match the reference.
// T = true  -> grid is channel-last (D,H,W,C), vectorized b128 gathers
// T = false -> original (C,D,H,W) layout, scalar gathers
// ---------------------------------------------------------------------------
template<int RESO, int C, bool T>
__device__ __forceinline__ void grid_feat(const float* __restrict__ g,
                                          float ppx, float ppy, float ppz,
                                          float* __restrict__ feat) {
  constexpr int R = RESO;
  constexpr bool POW2 = (R & (R - 1)) == 0;
  const float scale = 2.0f / (float)R;
  const float half = scale * 0.5f;
  float ix, iy, iz;
  {
#pragma clang fp contract(off)
    float c;
    if constexpr (POW2) c = ppx * (float)R; else c = ppx / half;
    c = c - 1.0f; c = fminf(fmaxf(c, -1.0f), 1.0f);
    ix = fminf(fmaxf((c + 1.0f) * 0.5f * (float)(R - 1), 0.0f), (float)(R - 1));
    if constexpr (POW2) c = ppy * (float)R; else c = ppy / half;
    c = c - 1.0f; c = fminf(fmaxf(c, -1.0f), 1.0f);
    iy = fminf(fmaxf((c + 1.0f) * 0.5f * (float)(R - 1), 0.0f), (float)(R - 1));
    if constexpr (POW2) c = ppz * (float)R; else c = ppz / half;
    c = c - 1.0f; c = fminf(fmaxf(c, -1.0f), 1.0f);
    iz = fminf(fmaxf((c + 1.0f) * 0.5f * (float)(R - 1), 0.0f), (float)(R - 1));
  }
  const float x0f = floorf(ix), y0f = floorf(iy), z0f = floorf(iz);
  const float wx = ix - x0f, wy = iy - y0f, wz = iz - z0f;
  const int x0 = (int)x0f, y0 = (int)y0f, z0 = (int)z0f;
  const int x1 = min(x0 + 1, R - 1), y1 = min(y0 + 1, R - 1), z1 = min(z0 + 1, R - 1);
  const float mx = 1.0f - wx, my = 1.0f - wy, mz = 1.0f - wz;
  const float w[8] = { mz*my*mx, mz*my*wx, mz*wy*mx, mz*wy*wx,
                       wz*my*mx, wz*my*wx, wz*wy*mx, wz*wy*wx };
  const int xs[2] = {x0, x1}, ys[2] = {y0, y1}, zs[2] = {z0, z1};
#pragma unroll
  for (int k = 0; k < 8; ++k) {
    const int xi = xs[k & 1], yi = ys[(k >> 1) & 1], zi = zs[(k >> 2) & 1];
    if constexpr (T) {
      const float* base = g + (size_t)(((zi * R) + yi) * R + xi) * C;
#pragma unroll
      for (int cc = 0; cc < C; cc += 4) {
        const v4f v = *reinterpret_cast<const v4f*>(base + cc);
        feat[cc + 0] += w[k] * v.x;
        feat[cc + 1] += w[k] * v.y;
        feat[cc + 2] += w[k] * v.z;
        feat[cc + 3] += w[k] * v.w;
      }
    } else {
      const size_t off = (size_t)((zi * R + yi) * R + xi);
      constexpr size_t N = (size_t)R * R * R;
#pragma unroll
      for (int cc = 0; cc < C; ++cc)
        feat[cc] += w[k] * g[(size_t)cc * N + off];
    }
  }
}

// ---------------------------------------------------------------------------
// Main encoder: one thread per point. The block's (256 x 3) coordinate slab is
// staged into LDS with CDNA5 async global->LDS b128 copies (ASYNCcnt path),
// then each thread gathers 6 grids and writes 40 floats with non-temporal
// b128 stores (keeps the ~188 MB of grids resident in the 192 MB L2).
// ---------------------------------------------------------------------------
template<bool T>
__global__ __launch_bounds__(256) void encode_kernel(
    const float* __restrict__ x,
    const float* __restrict__ g0, const float* __restrict__ g1,
    const float* __restrict__ g2, const float* __restrict__ g3,
    const float* __restrict__ g4, const float* __restrict__ g5,
    float* __restrict__ out, int B) {
  __shared__ float sx[256 * 3];
  const int tid = threadIdx.x;
  const int base = blockIdx.x * 256;
  const int count = min(256, B - base);
  const int nfloats = count * 3;

  {
    const float* src = x + (size_t)base * 3;
    const int nvec = nfloats >> 2;               // full 16-byte chunks
    if (tid < nvec) {
      const unsigned lds  = (unsigned)(size_t)(const void*)&sx[tid * 4];
      const unsigned goff = (unsigned)tid * 16u;
      asm volatile("global_load_async_to_lds_b128 %0, %1, %2"
                   :: "v"(lds), "v"(goff), "s"(src) : "memory");
    }
    const int rem = nfloats & 3;                  // tail dwords (0 for full blocks)
    if (tid < rem) {
      const int e = (nvec << 2) + tid;
      const unsigned lds  = (unsigned)(size_t)(const void*)&sx[e];
      const unsigned goff = (unsigned)e * 4u;
      asm volatile("global_load_async_to_lds_b32 %0, %1, %2"
                   :: "v"(lds), "v"(goff), "s"(src) : "memory");
    }
    asm volatile("s_wait_asynccnt 0" ::: "memory");
    __syncthreads();
  }

  const int i = base + tid;
  if (i >= B) return;

  const float px = sx[tid * 3 + 0];
  const float py = sx[tid * 3 + 1];
  const float pz = sx[tid * 3 + 2];

  float x32, x48, x64, x96, x128, x192;
  float y32, y48, y64, y96, y128, y192;
  float z32, z48, z64, z96, z128, z192;
  sawtooth_all(px, x32, x48, x64, x96, x128, x192);
  sawtooth_all(py, y32, y48, y64, y96, y128, y192);
  sawtooth_all(pz, z32, z48, z64, z96, z128, z192);

  float feat[40];
#pragma unroll
  for (int j = 0; j < 40; ++j) feat[j] = 0.0f;

  grid_feat< 32, 8, T>(g0, x32,  y32,  z32,  feat + 0);
  grid_feat< 48, 8, T>(g1, x48,  y48,  z48,  feat + 8);
  grid_feat< 64, 8, T>(g2, x64,  y64,  z64,  feat + 16);
  grid_feat< 96, 8, T>(g3, x96,  y96,  z96,  feat + 24);
  grid_feat<128, 4, T>(g4, x128, y128, z128, feat + 32);
  grid_feat<192, 4, T>(g5, x192, y192, z192, feat + 36);

  float* op = out + (size_t)i * 40;
#pragma unroll
  for (int j = 0; j < 40; j += 4) {
    v4f v = { feat[j], feat[j + 1], feat[j + 2], feat[j + 3] };
    __builtin_nontemporal_store(v, reinterpret_cast<v4f*>(op + j));
  }
}

// ---------------------------------------------------------------------------
static constexpr int kRes[6] = {32, 48, 64, 96, 128, 192};
static constexpr int kCh[6]  = {8, 8, 8, 8, 4, 4};

extern "C" void kernel_launch(void* const* d_in, const int* in_sizes, int n_in,
                              void* d_out, int out_size, void* d_ws, size_t ws_size,
                              hipStream_t stream) {
  (void)n_in; (void)out_size;
  const float* x = (const float*)d_in[0];
  const float* g[6];
  for (int i = 0; i < 6; ++i) g[i] = (const float*)d_in[1 + i];
  const int B = in_sizes[0] / 3;
  float* out = (float*)d_out;

  size_t offs[6];
  size_t tot = 0;
  for (int i = 0; i < 6; ++i) {
    offs[i] = tot;
    tot += (size_t)kCh[i] * kRes[i] * kRes[i] * kRes[i];
  }
  const bool useT = (ws_size >= tot * sizeof(float));
  const int blocks = (B + 255) / 256;

  if (useT) {
    float* w = (float*)d_ws;
    for (int i = 0; i < 6; ++i) {
      const int N = kRes[i] * kRes[i] * kRes[i];
      if (kCh[i] == 8) {
        constexpr int S = 256 / 8;
        transpose_kernel<8><<<(N + S - 1) / S, 256, 0, stream>>>(g[i], w + offs[i], N);
      } else {
        constexpr int S = 256 / 4;
        transpose_kernel<4><<<(N + S - 1) / S, 256, 0, stream>>>(g[i], w + offs[i], N);
      }
    }
    encode_kernel<true><<<blocks, 256, 0, stream>>>(
        x, w + offs[0], w + offs[1], w + offs[2], w + offs[3], w + offs[4], w + offs[5],
        out, B);
  } else {
    encode_kernel<false><<<blocks, 256, 0, stream>>>(
        x, g[0], g[1], g[2], g[3], g[4], g[5], out, B);
  }
}